// OptimizedAttention_83949430768213
// MI455X (gfx1250) — compile-verified
//
#include <hip/hip_runtime.h>
#include <hip/hip_bf16.h>

typedef float v2f __attribute__((ext_vector_type(2)));
typedef float v8f __attribute__((ext_vector_type(8)));

#define D_   2048
#define H_   16
#define KV_  8
#define G_   2
#define DK_  128
#define S_   2048
#define PS_  16

__device__ __forceinline__ v8f wmma_f32(v2f a, v2f b, v8f c) {
    return __builtin_amdgcn_wmma_f32_16x16x4_f32(false, a, false, b, (short)0, c, false, false);
}

__device__ __forceinline__ float rmax16(float v) {
    #pragma unroll
    for (int m = 1; m < 16; m <<= 1) v = fmaxf(v, __shfl_xor(v, m, 32));
    return v;
}
__device__ __forceinline__ float rsum16(float v) {
    #pragma unroll
    for (int m = 1; m < 16; m <<= 1) v += __shfl_xor(v, m, 32);
    return v;
}

// ---------------------------------------------------------------------------
// FP32 GEMM: C[M,N] = A[M,K] @ B[K,N].  N,K compile-time (imm offsets).
// One wave owns a 64x16 M-strip (4 C tiles).  Per 16-K panel: batch-load all
// operands (one big clause + one wait), then 16 WMMAs with the 4 accumulator
// chains interleaved for back-to-back matrix-pipe issue.
// ---------------------------------------------------------------------------
template<int N, int K>
__global__ __launch_bounds__(32) void gemm64x16_wmma(const float* __restrict__ A,
                                                     const float* __restrict__ B,
                                                     float* __restrict__ C) {
    const int lane = threadIdx.x;
    const int half = lane >> 4;          // 0: K=k,k+1   1: K=k+2,k+3
    const int l    = lane & 15;
    const int n0 = blockIdx.x * 16;
    const int m0 = blockIdx.y * 64;

    const float* Ab = A + (size_t)(m0 + l) * K + 2 * half;   // row m0+l, K-pair slot
    const float* Bb = B + (size_t)(2 * half) * N + n0 + l;   // col n0+l, K-pair slot

    v8f acc[4];
    #pragma unroll
    for (int i = 0; i < 4; ++i) acc[i] = (v8f){};

    for (int k0 = 0; k0 < K; k0 += 16, Ab += 16, Bb += (size_t)16 * N) {
        __builtin_prefetch(Ab + 64, 0, 0);                    // global_prefetch_b8
        __builtin_prefetch(Bb + (size_t)64 * N, 0, 0);
        v2f bt[4], at[4][4];
        #pragma unroll
        for (int c = 0; c < 4; ++c) {                          // batch all loads first
            bt[c].x = Bb[(size_t)(4 * c) * N];                 // B[k,   n] (imm offset)
            bt[c].y = Bb[(size_t)(4 * c + 1) * N];             // B[k+1, n]
            #pragma unroll
            for (int i = 0; i < 4; ++i)
                at[c][i] = *(const v2f*)(Ab + 4 * c + (size_t)i * 16 * K);
        }
        #pragma unroll
        for (int c = 0; c < 4; ++c)                            // then an uninterrupted
            #pragma unroll
            for (int i = 0; i < 4; ++i)                        // WMMA burst (i-inner:
                acc[i] = wmma_f32(at[c][i], bt[c], acc[i]);    // independent chains)
    }
    #pragma unroll
    for (int i = 0; i < 4; ++i) {
        float* Cp = C + (size_t)(m0 + 16 * i) * N + n0 + l;
        #pragma unroll
        for (int r = 0; r < 8; ++r)
            Cp[(size_t)(r + 8 * half) * N] = acc[i][r];        // C layout: M=r / M=r+8
    }
}

// ---------------------------------------------------------------------------
// RMSNorm + RoPE for Q, in place. One wave per (s, head); 4 elems/lane.
// ---------------------------------------------------------------------------
__global__ __launch_bounds__(32) void rmsrope_q(float* __restrict__ q,
                                                const float* __restrict__ gain) {
    const int s = blockIdx.x, h = blockIdx.y, lane = threadIdx.x;
    float* base = q + ((size_t)s * H_ + h) * DK_;
    float x1a = base[lane],      x1b = base[lane + 32];
    float x2a = base[lane + 64], x2b = base[lane + 96];
    float ss = x1a * x1a + x1b * x1b + x2a * x2a + x2b * x2b;
    #pragma unroll
    for (int m = 1; m < 32; m <<= 1) ss += __shfl_xor(ss, m, 32);
    const float r = rsqrtf(ss * (1.0f / 128.0f) + 1e-6f);
    float n1a = x1a * r * gain[lane];
    float n1b = x1b * r * gain[lane + 32];
    float n2a = x2a * r * gain[lane + 64];
    float n2b = x2b * r * gain[lane + 96];
    const float LN1E4 = 9.2103403719761827f;
    float fa = __expf(-(float)lane        * (1.0f / 64.0f) * LN1E4);
    float fb = __expf(-(float)(lane + 32) * (1.0f / 64.0f) * LN1E4);
    float ca, sa, cb, sb;
    __sincosf((float)s * fa, &sa, &ca);
    __sincosf((float)s * fb, &sb, &cb);
    base[lane]      = n1a * ca - n2a * sa;
    base[lane + 64] = n1a * sa + n2a * ca;
    base[lane + 32] = n1b * cb - n2b * sb;
    base[lane + 96] = n1b * sb + n2b * cb;
}

// ---------------------------------------------------------------------------
// RMSNorm + RoPE for K, scattered into the paged K cache (workspace copy).
// ---------------------------------------------------------------------------
__global__ __launch_bounds__(32) void rmsrope_k(const float* __restrict__ k,
                                                const float* __restrict__ gain,
                                                const int* __restrict__ positions,
                                                const int* __restrict__ page_table,
                                                float* __restrict__ kcache) {
    const int s = blockIdx.x, h = blockIdx.y, lane = threadIdx.x;
    const float* base = k + ((size_t)s * KV_ + h) * DK_;
    float x1a = base[lane],      x1b = base[lane + 32];
    float x2a = base[lane + 64], x2b = base[lane + 96];
    float ss = x1a * x1a + x1b * x1b + x2a * x2a + x2b * x2b;
    #pragma unroll
    for (int m = 1; m < 32; m <<= 1) ss += __shfl_xor(ss, m, 32);
    const float r = rsqrtf(ss * (1.0f / 128.0f) + 1e-6f);
    float n1a = x1a * r * gain[lane];
    float n1b = x1b * r * gain[lane + 32];
    float n2a = x2a * r * gain[lane + 64];
    float n2b = x2b * r * gain[lane + 96];
    const float LN1E4 = 9.2103403719761827f;
    float fa = __expf(-(float)lane        * (1.0f / 64.0f) * LN1E4);
    float fb = __expf(-(float)(lane + 32) * (1.0f / 64.0f) * LN1E4);
    float ca, sa, cb, sb;
    __sincosf((float)s * fa, &sa, &ca);
    __sincosf((float)s * fb, &sb, &cb);
    const int pos  = positions[s];
    const int phys = page_table[pos / PS_] * PS_ + pos % PS_;
    float* dst = kcache + ((size_t)h * S_ + phys) * DK_;
    dst[lane]      = n1a * ca - n2a * sa;
    dst[lane + 64] = n1a * sa + n2a * ca;
    dst[lane + 32] = n1b * cb - n2b * sb;
    dst[lane + 96] = n1b * sb + n2b * cb;
}

// ---------------------------------------------------------------------------
// V scatter through the CDNA5 async global<->LDS path: each lane DMAs 16B
// global->LDS, waits on ASYNCcnt, then DMAs LDS->global to the paged slot.
// ---------------------------------------------------------------------------
__global__ __launch_bounds__(32) void vscatter_async(const float* __restrict__ v,
                                                     const int* __restrict__ positions,
                                                     const int* __restrict__ page_table,
                                                     float* __restrict__ vcache) {
    __shared__ float buf[DK_];
    const int s = blockIdx.x, h = blockIdx.y, lane = threadIdx.x;
    const int pos  = positions[s];
    const int phys = page_table[pos / PS_] * PS_ + pos % PS_;
    const float* src = v + ((size_t)s * KV_ + h) * DK_ + lane * 4;
    float*       dst = vcache + ((size_t)h * S_ + phys) * DK_ + lane * 4;
    unsigned lds_off = (unsigned)(size_t)&buf[lane * 4];          // LDS byte offset
    unsigned long long ga = (unsigned long long)(size_t)src;
    unsigned long long gd = (unsigned long long)(size_t)dst;
    asm volatile("global_load_async_to_lds_b128 %0, %1, off"
                 :: "v"(lds_off), "v"(ga) : "memory");
    asm volatile("s_wait_asynccnt 0" ::: "memory");
    asm volatile("global_store_async_from_lds_b128 %0, %1, off"
                 :: "v"(gd), "v"(lds_off) : "memory");
    asm volatile("s_wait_asynccnt 0" ::: "memory");
}

// ---------------------------------------------------------------------------
// Flash attention (causal, GQA G=2). One wave per (16-query tile, head).
// Operand loads batched ahead of each WMMA burst (one clause + one wait).
// ---------------------------------------------------------------------------
__global__ __launch_bounds__(32) void attn16(const float* __restrict__ q,
                                             const float* __restrict__ kc,
                                             const float* __restrict__ vc,
                                             float* __restrict__ o) {
    const int qt = blockIdx.x, h = blockIdx.y;
    const int lane = threadIdx.x;
    const int half = lane >> 4;
    const int l    = lane & 15;
    const int kvh  = h / G_;
    __shared__ float Pl[16 * 20];

    const float SC = 0.08838834764831845f;     // 1/sqrt(DK)
    const float* Qb = q + ((size_t)(qt * 16 + l) * H_ + h) * DK_ + 2 * half;
    v2f qa[32];
    #pragma unroll
    for (int j = 0; j < 32; ++j) {
        v2f t = *(const v2f*)(Qb + 4 * j);
        qa[j].x = t.x * SC; qa[j].y = t.y * SC;
    }

    v8f acc[8];
    #pragma unroll
    for (int t = 0; t < 8; ++t) acc[t] = (v8f){};
    float mrun[8], lrun[8];
    #pragma unroll
    for (int r = 0; r < 8; ++r) { mrun[r] = -1e30f; lrun[r] = 0.0f; }

    const float* Kb = kc + (size_t)kvh * S_ * DK_;
    const float* Vb = vc + (size_t)kvh * S_ * DK_;

    for (int kb = 0; kb <= qt; ++kb) {
        // ---- scores S = Q K^T : 4 groups of (8 batched loads -> 8 WMMAs)
        v8f s = {};
        const float* Krow = Kb + (size_t)(kb * 16 + l) * DK_ + 2 * half;
        #pragma unroll
        for (int jj = 0; jj < 32; jj += 8) {
            v2f kt[8];
            #pragma unroll
            for (int u = 0; u < 8; ++u)
                kt[u] = *(const v2f*)(Krow + 4 * (jj + u));   // one load clause
            #pragma unroll
            for (int u = 0; u < 8; ++u)
                s = wmma_f32(qa[jj + u], kt[u], s);           // WMMA burst
        }
        if (kb == qt) {                             // causal mask, diagonal block only
            const int ki = kb * 16 + l;
            #pragma unroll
            for (int r = 0; r < 8; ++r)
                if (ki > qt * 16 + r + 8 * half) s[r] = -1e30f;
        }
        // ---- online softmax (row stats via 16-lane shuffle reductions)
        #pragma unroll
        for (int r = 0; r < 8; ++r) {
            float mn = fmaxf(mrun[r], rmax16(s[r]));
            float sc = __expf(mrun[r] - mn);
            float p  = __expf(s[r] - mn);
            lrun[r] = lrun[r] * sc + rsum16(p);
            mrun[r] = mn;
            #pragma unroll
            for (int t = 0; t < 8; ++t) acc[t][r] *= sc;
            Pl[(r + 8 * half) * 20 + l] = p;        // P row-major in padded LDS
        }
        __syncthreads();                            // single-wave WG: cheap DS wait
        // ---- acc += P V : per key-chunk, batch 8 V operands then 8 independent WMMAs
        #pragma unroll
        for (int j = 0; j < 4; ++j) {
            v2f a;
            a.x = Pl[l * 20 + 4 * j + 2 * half];
            a.y = Pl[l * 20 + 4 * j + 2 * half + 1];
            const float* Vrow = Vb + (size_t)(kb * 16 + 4 * j + 2 * half) * DK_ + l;
            v2f vt[8];
            #pragma unroll
            for (int t = 0; t < 8; ++t) {
                vt[t].x = Vrow[t * 16];             // V[key,   d]  (one load clause)
                vt[t].y = Vrow[DK_ + t * 16];       // V[key+1, d]
            }
            #pragma unroll
            for (int t = 0; t < 8; ++t)
                acc[t] = wmma_f32(a, vt[t], acc[t]); // independent accumulators
        }
        __syncthreads();
    }

    float inv[8];
    #pragma unroll
    for (int r = 0; r < 8; ++r) inv[r] = 1.0f / lrun[r];
    float* Op = o + (size_t)(qt * 16) * D_ + h * DK_ + l;
    #pragma unroll
    for (int t = 0; t < 8; ++t)
        #pragma unroll
        for (int r = 0; r < 8; ++r)
            Op[(size_t)(r + 8 * half) * D_ + t * 16] = acc[t][r] * inv[r];
}

// ---------------------------------------------------------------------------
extern "C" void kernel_launch(void* const* d_in, const int* in_sizes, int n_in,
                              void* d_out, int out_size, void* d_ws, size_t ws_size,
                              hipStream_t stream) {
    const float* x    = (const float*)d_in[0];
    const float* wq   = (const float*)d_in[1];
    const float* wk   = (const float*)d_in[2];
    const float* wv   = (const float*)d_in[3];
    const float* wo   = (const float*)d_in[4];
    const float* qg   = (const float*)d_in[5];
    const float* kg   = (const float*)d_in[6];
    const int*   pt   = (const int*)d_in[9];
    const int*   pos  = (const int*)d_in[10];
    float*       out  = (float*)d_out;

    float* qb = (float*)d_ws;                         // S*H*DK
    float* kb = qb + (size_t)S_ * H_ * DK_;           // S*KV*DK
    float* vb = kb + (size_t)S_ * KV_ * DK_;          // S*KV*DK
    float* kc = vb + (size_t)S_ * KV_ * DK_;          // KV*S*DK paged K cache
    float* vc = kc + (size_t)KV_ * S_ * DK_;          // KV*S*DK paged V cache
    float* ab = vc + (size_t)KV_ * S_ * DK_;          // S*D attention output

    // Projections (fp32 WMMA GEMMs, 64x16 per wave, compile-time strides)
    gemm64x16_wmma<H_ * DK_, D_><<<dim3((H_ * DK_) / 16, S_ / 64), 32, 0, stream>>>(x, wq, qb);
    gemm64x16_wmma<KV_ * DK_, D_><<<dim3((KV_ * DK_) / 16, S_ / 64), 32, 0, stream>>>(x, wk, kb);
    gemm64x16_wmma<KV_ * DK_, D_><<<dim3((KV_ * DK_) / 16, S_ / 64), 32, 0, stream>>>(x, wv, vb);

    // RMSNorm + RoPE (+ paged cache scatter for K/V)
    rmsrope_q<<<dim3(S_, H_), 32, 0, stream>>>(qb, qg);
    rmsrope_k<<<dim3(S_, KV_), 32, 0, stream>>>(kb, kg, pos, pt, kc);
    vscatter_async<<<dim3(S_, KV_), 32, 0, stream>>>(vb, pos, pt, vc);

    // Flash attention
    attn16<<<dim3(S_ / 16, H_), 32, 0, stream>>>(qb, kc, vc, ab);

    // Output projection
    gemm64x16_wmma<D_, D_><<<dim3(D_ / 16, S_ / 64), 32, 0, stream>>>(ab, wo, out);
}